// MultiGVPConvLayer_75419625718340
// MI455X (gfx1250) — compile-verified
//
#include <hip/hip_runtime.h>
#include <math.h>

#define NN   50000
#define EE   800000
#define SDIM 128
#define VDIM 16
#define V3   48
#define ESD  32

typedef float v2f __attribute__((ext_vector_type(2)));
typedef float v8f __attribute__((ext_vector_type(8)));

// D = A(16x4 f32) * B(4x16 f32) + C(16x16 f32), exact f32 matrix pipe on CDNA5.
__device__ __forceinline__ v8f wmma4(v2f a, v2f b, v8f c) {
  return __builtin_amdgcn_wmma_f32_16x16x4_f32(false, a, false, b, (short)0, c,
                                               false, false);
}

// hardware v_rcp_f32 (1 ulp) instead of IEEE division expansion
__device__ __forceinline__ float fast_rcp(float x) {
  return __builtin_amdgcn_rcpf(x);
}

__device__ __forceinline__ float sigmoidf_(float x) {
  return fast_rcp(1.f + __expf(-x));
}

__device__ __forceinline__ void fadd_atomic(float* p, float v) {
  unsafeAtomicAdd(p, v);   // lowers to global_atomic_add_f32 (L2 atomic units)
}

// ---------------------------------------------------------------- zero scratch
__global__ void k_zero(float* __restrict__ p, int n) {
  int i = blockIdx.x * blockDim.x + threadIdx.x;
  int stride = gridDim.x * blockDim.x;
  for (; i < n; i += stride) p[i] = 0.f;
}

// ------------------------------------------------- per-node LN + vector norm
__global__ void __launch_bounds__(256) k_norm(
    const float* __restrict__ node_s, const float* __restrict__ node_v,
    const float* __restrict__ ln_g, const float* __restrict__ ln_b,
    float* __restrict__ s_n, float* __restrict__ v_n) {
  int wave = (blockIdx.x * blockDim.x + threadIdx.x) >> 5;
  int lane = threadIdx.x & 31;
  if (wave >= NN) return;

  const float* srow = node_s + (long)wave * SDIM;
  float4 x = ((const float4*)srow)[lane];
  float s = x.x + x.y + x.z + x.w;
#pragma unroll
  for (int m = 16; m >= 1; m >>= 1) s += __shfl_xor(s, m, 32);
  float mu = s * (1.f / SDIM);
  float dx = x.x - mu, dy = x.y - mu, dz = x.z - mu, dw = x.w - mu;
  float q = dx * dx + dy * dy + dz * dz + dw * dw;
#pragma unroll
  for (int m = 16; m >= 1; m >>= 1) q += __shfl_xor(q, m, 32);
  float rs = rsqrtf(q * (1.f / SDIM) + 1e-5f);
  int c0 = lane * 4;
  float4 o;
  o.x = dx * rs * ln_g[c0 + 0] + ln_b[c0 + 0];
  o.y = dy * rs * ln_g[c0 + 1] + ln_b[c0 + 1];
  o.z = dz * rs * ln_g[c0 + 2] + ln_b[c0 + 2];
  o.w = dw * rs * ln_g[c0 + 3] + ln_b[c0 + 3];
  ((float4*)(s_n + (long)wave * SDIM))[lane] = o;

  // vector channels: vnorm = sqrt(sum(|v|^2)/VD + eps)
  const float* vrow = node_v + (long)wave * V3;
  float2 v = {0.f, 0.f};
  int vi = lane * 2;
  if (vi < V3) v = ((const float2*)vrow)[lane];
  float qq = v.x * v.x + v.y * v.y;
#pragma unroll
  for (int m = 16; m >= 1; m >>= 1) qq += __shfl_xor(qq, m, 32);
  float inv = rsqrtf(qq * (1.f / VDIM) + 1e-8f);
  if (vi < V3) {
    float2 ov;
    ov.x = v.x * inv;
    ov.y = v.y * inv;
    ((float2*)(v_n + (long)wave * V3))[lane] = ov;
  }
}

// -------------------------------------------- edge GVP + scatter-add (atomics)
// One wave handles 16 edges: A = edge_s[16x32], B = wes^T, 8 column tiles.
__global__ void __launch_bounds__(256) k_edge(
    const float* __restrict__ edge_s, const float* __restrict__ edge_v,
    const int* __restrict__ edge_dst,
    const float* __restrict__ wes, const float* __restrict__ bes,
    const float* __restrict__ wev, const float* __restrict__ bev,
    float* __restrict__ agg_s, float* __restrict__ agg_v,
    float* __restrict__ cnt) {
  __shared__ float lds_gate[8][16 * 16];
  int wid = threadIdx.x >> 5;
  int lane = threadIdx.x & 31;
  int group = blockIdx.x * 8 + wid;
  if (group >= EE / 16) return;
  int ebase = group * 16;
  int row = lane & 15;   // A-matrix M / B-matrix N index
  int half = lane >> 4;  // K-pair selector per ISA 16x4 layout

  const float* arow = edge_s + (long)(ebase + row) * ESD;

  int dsts[8];
#pragma unroll
  for (int r = 0; r < 8; r++) dsts[r] = edge_dst[ebase + r + 8 * half];

#pragma unroll 1
  for (int c = 0; c < 8; c++) {
    v8f acc = {};
    int n = 16 * c + row;
    const float* brow = wes + (long)n * ESD;
#pragma unroll
    for (int j = 0; j < 8; j++) {
      int k0 = 4 * j + 2 * half;
      v2f a = *(const v2f*)(arow + k0);
      v2f b = *(const v2f*)(brow + k0);
      acc = wmma4(a, b, acc);
    }
    float bias = bes[n];
#pragma unroll
    for (int r = 0; r < 8; r++) {
      float pre = acc[r] + bias;
      float so = pre * sigmoidf_(pre);  // SiLU
      if (c == 0) lds_gate[wid][(r + 8 * half) * 16 + n] = sigmoidf_(so);
      fadd_atomic(agg_s + (long)dsts[r] * SDIM + n, so);
    }
  }

  // per-edge incoming count
  if (lane < 16) fadd_atomic(cnt + edge_dst[ebase + lane], 1.f);

  // vector messages: EVD=1 -> tiny 48x3 matvec per edge, gated, scattered
  int e = ebase + row;
  int dstv = edge_dst[e];
  float ev0 = edge_v[e * 3 + 0];
  float ev1 = edge_v[e * 3 + 1];
  float ev2 = edge_v[e * 3 + 2];
#pragma unroll
  for (int t = 0; t < 24; t++) {
    int d = half * 24 + t;
    float v = ev0 * wev[d * 3 + 0] + ev1 * wev[d * 3 + 1] +
              ev2 * wev[d * 3 + 2] + bev[d];
    float g = lds_gate[wid][row * 16 + d / 3];
    fadd_atomic(agg_v + (long)dstv * V3 + d, v * g);
  }
}

// ------------------------------- node GVP + residual linears -> final outputs
// One wave handles 16 nodes; dual accumulators run GVP and residual GEMMs.
__global__ void __launch_bounds__(256) k_node(
    const float* __restrict__ s_n, const float* __restrict__ v_n,
    const float* __restrict__ agg_s, const float* __restrict__ agg_v,
    const float* __restrict__ cnt,
    const float* __restrict__ wns, const float* __restrict__ bns,
    const float* __restrict__ wnv, const float* __restrict__ bnv,
    const float* __restrict__ wrs, const float* __restrict__ brs,
    const float* __restrict__ wrv, const float* __restrict__ brv,
    float* __restrict__ out) {
  __shared__ float lds_gate[8][16 * 16];
  int wid = threadIdx.x >> 5;
  int lane = threadIdx.x & 31;
  int group = blockIdx.x * 8 + wid;
  if (group >= NN / 16) return;
  int nb = group * 16;
  int row = lane & 15;
  int half = lane >> 4;
  int noderow = nb + row;

  float invden = fast_rcp(fmaxf(cnt[noderow], 1.f));
  const float* sA = s_n + (long)noderow * SDIM;
  const float* gA = agg_s + (long)noderow * SDIM;

#pragma unroll 1
  for (int ct = 0; ct < 8; ct++) {
    v8f ao = {};
    v8f ar = {};
    int n = 16 * ct + row;
    const float* bo = wns + (long)n * SDIM;
    const float* br = wrs + (long)n * SDIM;
#pragma unroll 4
    for (int j = 0; j < 32; j++) {
      int k0 = 4 * j + 2 * half;
      v2f as = *(const v2f*)(sA + k0);
      v2f ag = *(const v2f*)(gA + k0);
      v2f ain = {as.x + ag.x * invden, as.y + ag.y * invden};
      v2f bO = *(const v2f*)(bo + k0);
      v2f bR = *(const v2f*)(br + k0);
      ao = wmma4(ain, bO, ao);
      ar = wmma4(as, bR, ar);
    }
    float biasO = bns[n], biasR = brs[n];
#pragma unroll
    for (int r = 0; r < 8; r++) {
      int m = r + 8 * half;
      float pre = ao[r] + biasO;
      float so = pre * sigmoidf_(pre);  // SiLU
      if (ct == 0) lds_gate[wid][m * 16 + n] = sigmoidf_(so);
      out[(long)(nb + m) * SDIM + n] = so + (ar[r] + biasR);
    }
  }

  const float* vA = v_n + (long)noderow * V3;
  const float* gV = agg_v + (long)noderow * V3;
  float* outv = out + (long)NN * SDIM;
#pragma unroll 1
  for (int ct = 0; ct < 3; ct++) {
    v8f ao = {};
    v8f ar = {};
    int d = 16 * ct + row;
    const float* bo = wnv + (long)d * V3;
    const float* br = wrv + (long)d * V3;
#pragma unroll
    for (int j = 0; j < 12; j++) {
      int k0 = 4 * j + 2 * half;
      v2f av = *(const v2f*)(vA + k0);
      v2f ag = *(const v2f*)(gV + k0);
      v2f ain = {av.x + ag.x * invden, av.y + ag.y * invden};
      v2f bO = *(const v2f*)(bo + k0);
      v2f bR = *(const v2f*)(br + k0);
      ao = wmma4(ain, bO, ao);
      ar = wmma4(av, bR, ar);
    }
    float biasO = bnv[d], biasR = brv[d];
    int vo = d / 3;
#pragma unroll
    for (int r = 0; r < 8; r++) {
      int m = r + 8 * half;
      float pre = ao[r] + biasO;
      float g = lds_gate[wid][m * 16 + vo];
      outv[(long)(nb + m) * V3 + d] = pre * g + (ar[r] + biasR);
    }
  }
}

extern "C" void kernel_launch(void* const* d_in, const int* in_sizes, int n_in,
                              void* d_out, int out_size, void* d_ws,
                              size_t ws_size, hipStream_t stream) {
  const float* node_s = (const float*)d_in[0];
  const float* node_v = (const float*)d_in[1];
  const int* eidx     = (const int*)d_in[2];
  const float* edge_s = (const float*)d_in[3];
  const float* edge_v = (const float*)d_in[4];
  const float* ln_g = (const float*)d_in[5];
  const float* ln_b = (const float*)d_in[6];
  const float* wes = (const float*)d_in[7];
  const float* bes = (const float*)d_in[8];
  const float* wev = (const float*)d_in[9];
  const float* bev = (const float*)d_in[10];
  const float* wns = (const float*)d_in[11];
  const float* bns = (const float*)d_in[12];
  const float* wnv = (const float*)d_in[13];
  const float* bnv = (const float*)d_in[14];
  const float* wrs = (const float*)d_in[15];
  const float* brs = (const float*)d_in[16];
  const float* wrv = (const float*)d_in[17];
  const float* brv = (const float*)d_in[18];
  float* out = (float*)d_out;

  float* ws = (float*)d_ws;
  float* s_n = ws;                               // N*128
  float* v_n = s_n + (size_t)NN * SDIM;          // N*48
  float* agg_s = v_n + (size_t)NN * V3;          // N*128
  float* agg_v = agg_s + (size_t)NN * SDIM;      // N*48
  float* cnt = agg_v + (size_t)NN * V3;          // N
  const int* edge_dst = eidx + EE;               // edge_index[1]

  int zn = NN * (SDIM + V3 + 1);
  k_zero<<<2048, 256, 0, stream>>>(agg_s, zn);
  k_norm<<<(NN * 32 + 255) / 256, 256, 0, stream>>>(node_s, node_v, ln_g, ln_b,
                                                    s_n, v_n);
  k_edge<<<EE / 16 / 8, 256, 0, stream>>>(edge_s, edge_v, edge_dst, wes, bes,
                                          wev, bev, agg_s, agg_v, cnt);
  k_node<<<(NN / 16 + 7) / 8, 256, 0, stream>>>(s_n, v_n, agg_s, agg_v, cnt,
                                                wns, bns, wnv, bnv, wrs, brs,
                                                wrv, brv, out);
}